// Net_29935922053209
// MI455X (gfx1250) — compile-verified
//
#include <hip/hip_runtime.h>

#define N_NODES 100000
#define N_EDGES 3200000
#define F_IN    1433
#define HID     16
#define N_CLS   7

typedef float v2f __attribute__((ext_vector_type(2)));
typedef float v8f __attribute__((ext_vector_type(8)));

// ---------------------------------------------------------------------------
// GEMM1: t1 = x @ W1  (x: [N,1433] f32, W1: [1433,16] f32, t1: [N,16])
// One wave32 per 16-row tile. V_WMMA_F32_16X16X4_F32, K stepped by 4.
// A (16x4 f32) lane layout (ISA 7.12.2): lanes 0-15 = M rows, VGPR0/1 = K0/K1;
// lanes 16-31 = same M rows, VGPR0/1 = K2/K3  -> per-lane float2 load.
// B (4x16 f32): row (fixed K) striped over N in lanes; VGPR0 carries K=0(|2),
// VGPR1 carries K=1(|3) for lane half 0(|1).
// Also writes agg1 = t1 (self-loop contribution of the aggregation).
// ---------------------------------------------------------------------------
__global__ __launch_bounds__(256) void gemm1_wmma(const float* __restrict__ x,
                                                  const float* __restrict__ W1,
                                                  float* __restrict__ t1,
                                                  float* __restrict__ agg1) {
  const int wave = (int)((blockIdx.x * blockDim.x + threadIdx.x) >> 5);
  const int lane = (int)(threadIdx.x & 31);
  const int row0 = wave * 16;
  if (row0 >= N_NODES) return;

  const int half = lane >> 4;      // 0: lanes 0-15, 1: lanes 16-31
  const int l16  = lane & 15;
  const int koff = half * 2;       // this lane covers K = k+koff, k+koff+1
  const long arow = (long)(row0 + l16) * F_IN;

  v8f c = {};
  constexpr int KFULL = (F_IN / 4) * 4;   // 1432
  for (int k = 0; k < KFULL; k += 4) {
    v2f a = *(const v2f*)(x + arow + k + koff);            // global_load_b64
    v2f b;
    b.x = W1[(k + koff)     * HID + l16];
    b.y = W1[(k + koff + 1) * HID + l16];
    c = __builtin_amdgcn_wmma_f32_16x16x4_f32(false, a, false, b,
                                              (short)0, c, false, false);
  }
  // K remainder (single column k = 1432), zero padded
  {
    v2f a = {0.f, 0.f};
    v2f b = {0.f, 0.f};
    if (half == 0) {
      a.x = x[arow + KFULL];
      b.x = W1[KFULL * HID + l16];
    }
    c = __builtin_amdgcn_wmma_f32_16x16x4_f32(false, a, false, b,
                                              (short)0, c, false, false);
  }
  // C/D layout: VGPR r -> row = r + half*8, col = l16
#pragma unroll
  for (int r = 0; r < 8; ++r) {
    const long o = (long)(row0 + r + half * 8) * HID + l16;
    t1[o]   = c[r];
    agg1[o] = c[r];
  }
}

// ---------------------------------------------------------------------------
// Scatter-add over edges, 16 features per edge: agg[dst] += t[src]
// ---------------------------------------------------------------------------
__global__ void scatter16(const int* __restrict__ src, const int* __restrict__ dst,
                          const float* __restrict__ t, float* __restrict__ agg) {
  const long idx = (long)blockIdx.x * blockDim.x + threadIdx.x;
  if (idx >= (long)N_EDGES * HID) return;
  const int e = (int)(idx >> 4);
  const int f = (int)(idx & 15);
  const float v = t[(long)src[e] * HID + f];
  atomicAdd(&agg[(long)dst[e] * HID + f], v);
}

// h = relu(agg1 + b1)
__global__ void relu_bias(const float* __restrict__ agg, const float* __restrict__ b,
                          float* __restrict__ h) {
  const long idx = (long)blockIdx.x * blockDim.x + threadIdx.x;
  if (idx >= (long)N_NODES * HID) return;
  const float v = agg[idx] + b[idx & 15];
  h[idx] = v > 0.f ? v : 0.f;
}

// ---------------------------------------------------------------------------
// GEMM2: t2 = h @ W2  (h: [N,16], W2: [16,7]) using the same WMMA tile,
// with B columns >= 7 zero-padded. Writes t2 and agg2 = t2.
// ---------------------------------------------------------------------------
__global__ __launch_bounds__(256) void gemm2_wmma(const float* __restrict__ h,
                                                  const float* __restrict__ W2,
                                                  float* __restrict__ t2,
                                                  float* __restrict__ agg2) {
  const int wave = (int)((blockIdx.x * blockDim.x + threadIdx.x) >> 5);
  const int lane = (int)(threadIdx.x & 31);
  const int row0 = wave * 16;
  if (row0 >= N_NODES) return;

  const int half = lane >> 4;
  const int l16  = lane & 15;
  const int koff = half * 2;
  const long arow = (long)(row0 + l16) * HID;

  v8f c = {};
#pragma unroll
  for (int k = 0; k < HID; k += 4) {
    v2f a = *(const v2f*)(h + arow + k + koff);
    v2f b = {0.f, 0.f};
    if (l16 < N_CLS) {
      b.x = W2[(k + koff)     * N_CLS + l16];
      b.y = W2[(k + koff + 1) * N_CLS + l16];
    }
    c = __builtin_amdgcn_wmma_f32_16x16x4_f32(false, a, false, b,
                                              (short)0, c, false, false);
  }
  if (l16 < N_CLS) {
#pragma unroll
    for (int r = 0; r < 8; ++r) {
      const long o = (long)(row0 + r + half * 8) * N_CLS + l16;
      t2[o]   = c[r];
      agg2[o] = c[r];
    }
  }
}

// agg2[dst] += t2[src], 7 classes per edge
__global__ void scatter7(const int* __restrict__ src, const int* __restrict__ dst,
                         const float* __restrict__ t, float* __restrict__ agg) {
  const long idx = (long)blockIdx.x * blockDim.x + threadIdx.x;
  if (idx >= (long)N_EDGES * N_CLS) return;
  const int e = (int)(idx / N_CLS);
  const int f = (int)(idx - (long)e * N_CLS);
  const float v = t[(long)src[e] * N_CLS + f];
  atomicAdd(&agg[(long)dst[e] * N_CLS + f], v);
}

// out = agg2 + b2
__global__ void add_bias(const float* __restrict__ agg, const float* __restrict__ b,
                         float* __restrict__ out) {
  const long idx = (long)blockIdx.x * blockDim.x + threadIdx.x;
  if (idx >= (long)N_NODES * N_CLS) return;
  const int f = (int)(idx % N_CLS);
  out[idx] = agg[idx] + b[f];
}

extern "C" void kernel_launch(void* const* d_in, const int* in_sizes, int n_in,
                              void* d_out, int out_size, void* d_ws, size_t ws_size,
                              hipStream_t stream) {
  const float* x   = (const float*)d_in[0];
  const int*   ei  = (const int*)d_in[1];          // [2, N_EDGES]
  const float* W1  = (const float*)d_in[2];
  const float* b1  = (const float*)d_in[3];
  const float* W2  = (const float*)d_in[4];
  const float* b2  = (const float*)d_in[5];
  float*       out = (float*)d_out;

  const int* src = ei;
  const int* dst = ei + N_EDGES;

  // workspace layout (floats)
  float* ws    = (float*)d_ws;
  float* t1    = ws;                                   // N*16
  float* agg1  = t1   + (long)N_NODES * HID;           // N*16
  float* h     = agg1 + (long)N_NODES * HID;           // N*16
  float* t2    = h    + (long)N_NODES * HID;           // N*7
  float* agg2  = t2   + (long)N_NODES * N_CLS;         // N*7

  const int TPB = 256;
  const int tiles = (N_NODES + 15) / 16;               // 6250 waves
  const int gemmBlocks = (tiles * 32 + TPB - 1) / TPB;

  // Layer 1
  gemm1_wmma<<<gemmBlocks, TPB, 0, stream>>>(x, W1, t1, agg1);
  {
    const long n = (long)N_EDGES * HID;
    scatter16<<<(int)((n + TPB - 1) / TPB), TPB, 0, stream>>>(src, dst, t1, agg1);
  }
  {
    const long n = (long)N_NODES * HID;
    relu_bias<<<(int)((n + TPB - 1) / TPB), TPB, 0, stream>>>(agg1, b1, h);
  }

  // Layer 2
  gemm2_wmma<<<gemmBlocks, TPB, 0, stream>>>(h, W2, t2, agg2);
  {
    const long n = (long)N_EDGES * N_CLS;
    scatter7<<<(int)((n + TPB - 1) / TPB), TPB, 0, stream>>>(src, dst, t2, agg2);
  }
  {
    const long n = (long)N_NODES * N_CLS;
    add_bias<<<(int)((n + TPB - 1) / TPB), TPB, 0, stream>>>(agg2, b2, out);
  }
}